// LSTMAdvanced_26173530702473
// MI455X (gfx1250) — compile-verified
//
#include <hip/hip_runtime.h>
#include <hip/hip_bf16.h>
#include <stdint.h>

// ---------------------------------------------------------------------------
// LSTM (B=32, S=2048, E=512, H=512) for MI455X / gfx1250 (wave32, WMMA).
//
// Kernel 1: x_proj = x @ W_ih.T + (b_ih+b_hh)  via V_WMMA_F32_16X16X4_F32
//           (fp32 tensor op keeps reference precision; phase is BW-bound:
//           ~670MB traffic @ 23.3TB/s ~= 29us, 137 GFLOP easily hidden).
// Kernel 2: persistent 32-WG scan. Each WG owns 16 h-dims; its 64 rows of
//           W_hh (4 gates x 16 dims x 512 K x 4B = 128KB) live in LDS for the
//           whole scan, loaded once by the Tensor Data Mover with LDS padding
//           (pitch 516 floats -> conflict-free, 8B-aligned ds_load_b64 B
//           fragments). Per step: 8 waves x 1 WMMA tile (16x16, K=512), gate
//           exchange via LDS, fused cell + output masking + hidden capture,
//           then an agent-scope atomic grid barrier (h double-buffered in L2).
// ---------------------------------------------------------------------------

typedef float v2f __attribute__((ext_vector_type(2)));
typedef float v8f __attribute__((ext_vector_type(8)));
typedef unsigned int v4u __attribute__((ext_vector_type(4)));
typedef int v4i __attribute__((ext_vector_type(4)));
typedef int v8i __attribute__((ext_vector_type(8)));

#define LB   32
#define LS   2048
#define LE   512
#define LH   512
#define LG   2048          // 4H
#define NWG  32            // persistent scan workgroups (one per 16 h-dims)
#define WPITCH 516         // LDS row pitch in floats (512 + 2 TDM pads of 2)

__device__ __forceinline__ float sigmoidf_(float x) {
  return 1.0f / (1.0f + __expf(-x));
}

// ---------------------------------------------------------------------------
// Kernel 1: x_proj[m, n] = sum_k x[m,k] * W_ih[n,k] + b_ih[n] + b_hh[n]
//   M = B*S = 65536, N = 4H = 2048, K = E = 512.
//   Block = 256 threads (8 waves) -> tile M=32 (2 waves) x N=256 (4 waves).
//   Each wave: 16x64 C (4 accumulators), K loop of 4 with f32 WMMA.
// ---------------------------------------------------------------------------
__global__ __launch_bounds__(256) void xproj_gemm(
    const float* __restrict__ x, const float* __restrict__ W_ih,
    const float* __restrict__ b_ih, const float* __restrict__ b_hh,
    float* __restrict__ xproj) {
  const int lane = threadIdx.x & 31;
  const int w    = threadIdx.x >> 5;
  const int mi   = w & 1;
  const int ni   = w >> 1;
  const int m0   = blockIdx.x * 32 + mi * 16;
  const int n0   = blockIdx.y * 256 + ni * 64;
  const int lm   = lane & 15;       // M (A) / N (B,C) index within tile
  const int lh   = lane >> 4;       // lane half selects K pair / M offset

  v8f acc[4];
#pragma unroll
  for (int nt = 0; nt < 4; ++nt) acc[nt] = (v8f){0,0,0,0,0,0,0,0};

  const float* arow = x + (size_t)(m0 + lm) * LE + 2 * lh;
  for (int k0 = 0; k0 < LE; k0 += 4) {
    v2f a = *(const v2f*)(arow + k0);
#pragma unroll
    for (int nt = 0; nt < 4; ++nt) {
      const int n = n0 + nt * 16 + lm;
      v2f b = *(const v2f*)(W_ih + (size_t)n * LE + k0 + 2 * lh);
      acc[nt] = __builtin_amdgcn_wmma_f32_16x16x4_f32(
          false, a, false, b, (short)0, acc[nt], false, false);
    }
  }

#pragma unroll
  for (int nt = 0; nt < 4; ++nt) {
    const int n = n0 + nt * 16 + lm;
    const float bias = b_ih[n] + b_hh[n];
#pragma unroll
    for (int r = 0; r < 8; ++r) {
      const int m = m0 + r + 8 * lh;                 // C: VGPR r -> M=r / r+8
      xproj[(size_t)m * LG + n] = acc[nt][r] + bias;
    }
  }
}

// ---------------------------------------------------------------------------
// TDM 2D tile load: tile_h rows x tile_w f32 elements, row stride in elements,
// with LDS padding of 2 dwords every 256 dwords (-> pitch 516 per 512-row).
// 6-arg builtin (this toolchain): (v4u, v8i, v4i, v4i, v8i, i32 cpol).
// ---------------------------------------------------------------------------
__device__ __forceinline__ void tdm_load_2d(const float* gsrc, unsigned lds_off,
                                            unsigned tile_w, unsigned tile_h,
                                            unsigned row_stride) {
  const unsigned long long ga = (unsigned long long)(uintptr_t)gsrc;
  v4u g0;
  g0.x = 1u;                                        // count=1, user D#
  g0.y = lds_off;                                   // lds_addr (bytes)
  g0.z = (unsigned)(ga & 0xFFFFFFFFu);              // global_addr[31:0]
  g0.w = (unsigned)((ga >> 32) & 0x01FFFFFFu) | (2u << 30);  // addr[56:32]|type=2
  v8i g1;
  g1[0] = (int)((2u << 16)      // data_size = 4B
              | (1u << 20)      // pad_enable
              | (7u << 22)      // pad_interval: every 256 dwords
              | (1u << 25));    // pad_amount: 2 dwords
  g1[1] = (int)((tile_w & 0xFFFFu) << 16);                       // tensor_dim0 lo
  g1[2] = (int)(((tile_w >> 16) & 0xFFFFu) | ((tile_h & 0xFFFFu) << 16));
  g1[3] = (int)(((tile_h >> 16) & 0xFFFFu) | ((tile_w & 0xFFFFu) << 16)); // tile_dim0
  g1[4] = (int)(tile_h & 0xFFFFu);                               // tile_dim1
  g1[5] = (int)row_stride;                                       // dim0_stride lo
  g1[6] = 0;
  g1[7] = 0;
  v4i g2 = (v4i){0, 0, 0, 0};
  v4i g3 = (v4i){0, 0, 0, 0};
  v8i g4 = (v8i){0, 0, 0, 0, 0, 0, 0, 0};
  __builtin_amdgcn_tensor_load_to_lds(g0, g1, g2, g3, g4, 0);
}

// ---------------------------------------------------------------------------
// Kernel 2: persistent scan. grid = 32 blocks x 256 threads; requires all 32
// blocks co-resident (MI455X has >> 32 WGPs; 139KB dyn-LDS -> 1 block/WGP).
// Wave w: m = w&1 (batch half), q = w>>1 (gate). C tile = gates for
// batch m*16..+16, gate q, h-dims j0..j0+16.
// ---------------------------------------------------------------------------
__global__ __launch_bounds__(256) void lstm_scan(
    const float* __restrict__ xproj, const float* __restrict__ W_hh,
    const int* __restrict__ lengths, float* __restrict__ out,
    float* __restrict__ hidden, float* __restrict__ h0buf,
    float* __restrict__ h1buf, int* __restrict__ counter) {
  extern __shared__ char smem[];
  float* wsl   = (float*)smem;                       // 64 rows x 516 floats
  float* gates = (float*)(smem + 64 * WPITCH * 4);   // [4][32][16]
  float* cst   = gates + 4 * 32 * 16;                // [32][16] cell state

  const int tid  = threadIdx.x;
  const int wg   = blockIdx.x;
  const int j0   = wg * 16;
  const int lane = tid & 31;
  const int w    = tid >> 5;
  const int m    = w & 1;
  const int q    = w >> 1;
  const int lm   = lane & 15;
  const int lh   = lane >> 4;

  // --- stage this WG's W_hh slice into LDS with the Tensor Data Mover ---
  if (w == 0) {
#pragma unroll
    for (int qq = 0; qq < 4; ++qq)
      tdm_load_2d(W_hh + (size_t)(qq * LH + j0) * LH,
                  (unsigned)(qq * 16 * WPITCH * 4), LH, 16, LH);
    __builtin_amdgcn_s_wait_tensorcnt(0);
  }
  cst[tid % 512] = 0.0f;              // tid<256: covers [0,256)
  cst[256 + (tid % 256)] = 0.0f;      // covers [256,512)
  __syncthreads();

  const float* wl = wsl + (q * 16 + lm) * WPITCH;

  for (int t = 0; t < LS; ++t) {
    const float* hprev = (t & 1) ? h1buf : h0buf;
    float*       hnext = (t & 1) ? h0buf : h1buf;

    // ---- gates(t) = h(t-1) @ W_hh.T : one 16x16 WMMA tile per wave ----
    v8f acc = (v8f){0, 0, 0, 0, 0, 0, 0, 0};
    const float* ha = hprev + (m * 16 + lm) * LH + 2 * lh;
    for (int k0 = 0; k0 < LH; k0 += 4) {
      const int k = k0 + 2 * lh;
      v2f a = *(const v2f*)(ha + k0);
      v2f b = *(const v2f*)(wl + k + 2 * (k >> 8));   // skip TDM pad dwords
      acc = __builtin_amdgcn_wmma_f32_16x16x4_f32(
          false, a, false, b, (short)0, acc, false, false);
    }
#pragma unroll
    for (int r = 0; r < 8; ++r) {
      const int bb = m * 16 + r + 8 * lh;             // C: VGPR r -> M=r / r+8
      gates[(q * 32 + bb) * 16 + lm] = acc[r];
    }
    __syncthreads();

    // ---- cell update: 512 elems (32 batch x 16 h-dims), 2 per thread ----
#pragma unroll
    for (int e = 0; e < 2; ++e) {
      const int idx = tid + e * 256;
      const int j   = idx & 15;
      const int bb  = idx >> 4;
      const float* xp = xproj + ((size_t)bb * LS + t) * LG + j0 + j;
      const float gi = gates[(0 * 32 + bb) * 16 + j] + xp[0 * LH];
      const float gf = gates[(1 * 32 + bb) * 16 + j] + xp[1 * LH];
      const float gg = gates[(2 * 32 + bb) * 16 + j] + xp[2 * LH];
      const float go = gates[(3 * 32 + bb) * 16 + j] + xp[3 * LH];
      const float i_ = sigmoidf_(gi);
      const float f_ = sigmoidf_(gf);
      const float g_ = tanhf(gg);
      const float o_ = sigmoidf_(go);
      const float cc = f_ * cst[idx] + i_ * g_;
      cst[idx] = cc;
      const float hh = o_ * tanhf(cc);
      hnext[bb * LH + j0 + j] = hh;
      const int len = lengths[bb];
      out[((size_t)bb * LS + t) * LH + j0 + j] = (t < len) ? hh : 0.0f;
      if (t == len - 1) hidden[bb * LH + j0 + j] = hh;
    }

    // ---- grid barrier: h(t) must be device-visible before step t+1 ----
    __threadfence();
    __syncthreads();
    if (tid == 0) {
      __hip_atomic_fetch_add(counter, 1, __ATOMIC_RELEASE,
                             __HIP_MEMORY_SCOPE_AGENT);
      const int target = NWG * (t + 1);
      while (__hip_atomic_load(counter, __ATOMIC_ACQUIRE,
                               __HIP_MEMORY_SCOPE_AGENT) < target)
        __builtin_amdgcn_s_sleep(1);
    }
    __syncthreads();
  }
}

__global__ void init_state(int* counter, float* h0, float* h1) {
  const int i = blockIdx.x * blockDim.x + threadIdx.x;
  if (i == 0) *counter = 0;
  if (i < LB * LH) { h0[i] = 0.0f; h1[i] = 0.0f; }
}

// ---------------------------------------------------------------------------
extern "C" void kernel_launch(void* const* d_in, const int* in_sizes, int n_in,
                              void* d_out, int out_size, void* d_ws,
                              size_t ws_size, hipStream_t stream) {
  const float* x       = (const float*)d_in[0];
  const int*   lengths = (const int*)d_in[1];
  const float* W_ih    = (const float*)d_in[2];
  const float* W_hh    = (const float*)d_in[3];
  const float* b_ih    = (const float*)d_in[4];
  const float* b_hh    = (const float*)d_in[5];

  // d_out = [out: B*S*H f32][lengths: B i32 (bit-identical 4B)][hidden: B*H f32]
  float* out       = (float*)d_out;
  char*  len_out   = (char*)d_out + (size_t)LB * LS * LH * 4;
  float* hidden    = (float*)d_out + (size_t)LB * LS * LH + LB;

  // workspace: [counter][h0][h1][xproj 512MB]
  char*  ws      = (char*)d_ws;
  int*   counter = (int*)ws;
  float* h0buf   = (float*)(ws + 256);
  float* h1buf   = (float*)(ws + 256 + (size_t)LB * LH * 4);
  float* xproj   = (float*)(ws + 256 + 2 * (size_t)LB * LH * 4);

  init_state<<<64, 256, 0, stream>>>(counter, h0buf, h1buf);

  // M=65536 / 32 = 2048 blocks in x, N=2048 / 256 = 8 in y
  xproj_gemm<<<dim3(2048, 8), 256, 0, stream>>>(x, W_ih, b_ih, b_hh, xproj);

  (void)hipMemcpyAsync(len_out, lengths, LB * sizeof(int),
                       hipMemcpyDeviceToDevice, stream);

  const size_t smem = (size_t)(64 * WPITCH + 4 * 32 * 16 + 32 * 16) * 4;
  lstm_scan<<<NWG, 256, smem, stream>>>(xproj, W_hh, lengths, out, hidden,
                                        h0buf, h1buf, counter);
}